// VQVAE_68375879352398
// MI455X (gfx1250) — compile-verified
//
#include <hip/hip_runtime.h>

// ---------------------------------------------------------------------------
// VQ-VAE forward for MI455X (gfx1250, wave32).
//
// Roofline: ~2.4 TFLOP fp32 vs ~3.3 GB fp32 activation traffic -> memory
// bound at 23.3 TB/s (~150us mem vs ~15us of V_WMMA_F32_16X16X4_F32 work).
// So: fp32 end-to-end, WMMA for every GEMM, effort on data movement:
//   * NHWC intermediates -> contiguous float2 im2col fragment loads
//   * weights pre-packed k-major (per parity class for transposed convs)
//   * compile-time geometry -> pure shift/mask 32-bit indexing
//   * NEW: Tensor Data Mover stages weight tiles (and the VQ codebook) into
//     LDS, double-buffered on TENSORcnt, with TDM padding for conflict-free
//     LDS reads.  Removes the 8x per-wave redundant weight fetch per block.
// ---------------------------------------------------------------------------

typedef float v2f __attribute__((ext_vector_type(2)));
typedef float v8f __attribute__((ext_vector_type(8)));
typedef unsigned v4u __attribute__((ext_vector_type(4)));
typedef int v4i __attribute__((ext_vector_type(4)));
typedef int v8i __attribute__((ext_vector_type(8)));

// A 16x4 f32 fragment (2 VGPRs): lanes 0-15 hold M=lane, K=k0,k0+1;
// lanes 16-31 hold M=lane-16, K=k0+2,k0+3.  B 4x16 mirrors it.
// C/D 16x16 f32 (8 VGPRs): vgpr r, lane<16 -> (M=r,N=lane); lane>=16 ->
// (M=r+8, N=lane-16).
__device__ __forceinline__ v8f wmma4(v2f a, v2f b, v8f c) {
  return __builtin_amdgcn_wmma_f32_16x16x4_f32(
      /*neg_a=*/false, a, /*neg_b=*/false, b,
      /*c_mod=*/(short)0, c, /*reuse_a=*/false, /*reuse_b=*/false);
}

// ---------------------------------------------------------------------------
// TDM: 2D tile (rows x tile_k f32 elements) global -> LDS.  D# per ISA §8.3/8.4:
// group0: count=1, lds_addr, global_addr[56:0], type=2.  group1: data_size=4B,
// optional LDS padding, tensor dims == tile dims (tile always in-bounds),
// tensor_dim0_stride = global row stride.  Issued by one wave; completion via
// s_wait_tensorcnt.
// ---------------------------------------------------------------------------
__device__ __forceinline__ void tdm_load_2d(unsigned lds_byte_off,
                                            const float* gsrc,
                                            unsigned tile_k, unsigned rows,
                                            unsigned row_stride_elems,
                                            unsigned pad_interval_code,
                                            unsigned pad_amount_code,
                                            bool pad_en) {
  const unsigned long long ga = (unsigned long long)(__SIZE_TYPE__)gsrc;
  v4u g0;
  g0[0] = 1u;                                   // count=1, user D#, no gather
  g0[1] = lds_byte_off;                         // lds_addr
  g0[2] = (unsigned)ga;                         // global_addr[31:0]
  g0[3] = (unsigned)((ga >> 32) & 0x01ffffffu)  // global_addr[56:32]
          | (2u << 30);                         // type=2 ("image")
  v8i g1;
  unsigned f0 = (2u << 16);  // wg_mask=0, data_size=2 (4 bytes)
  if (pad_en) f0 |= (1u << 20) | (pad_interval_code << 22) | (pad_amount_code << 25);
  g1[0] = (int)f0;
  g1[1] = (int)((tile_k & 0xffffu) << 16);                    // tensor_dim0 lo
  g1[2] = (int)((tile_k >> 16) | ((rows & 0xffffu) << 16));   // d0 hi | d1 lo
  g1[3] = (int)((rows >> 16) | ((tile_k & 0xffffu) << 16));   // d1 hi | tile_dim0
  g1[4] = (int)(rows & 0xffffu);                              // tile_dim1 (dim2=0)
  g1[5] = (int)row_stride_elems;                              // dim0_stride[31:0]
  g1[6] = 0;
  g1[7] = 0;
  v4i z4 = {};
#if defined(__clang_major__) && __clang_major__ >= 23
  v8i z8 = {};
  __builtin_amdgcn_tensor_load_to_lds(g0, g1, z4, z4, z8, 0);
#else
  __builtin_amdgcn_tensor_load_to_lds(g0, g1, z4, z4, 0);
#endif
}

__device__ __forceinline__ void tdm_wait0() {
  __builtin_amdgcn_s_wait_tensorcnt(0);
}

__device__ __forceinline__ unsigned lds_off(const void* p) {
  return (unsigned)(__SIZE_TYPE__)p;  // low 32 bits of LDS aperture = offset
}

// ---------------------------------------------------------------------------
// Weight repacks.
// conv: [COUT][CIN][KHW] -> [COUT][KHW][CIN]   (k = tap*CIN + ci)
// ---------------------------------------------------------------------------
template <int CIN, int KHW>
__global__ __launch_bounds__(256) void pack_conv_w(
    const float* __restrict__ w, float* __restrict__ out, unsigned total) {
  const unsigned o = blockIdx.x * 256u + threadIdx.x;
  if (o >= total) return;
  const unsigned ci = o % CIN;
  const unsigned r = o / CIN;
  const unsigned tap = r % KHW;
  const unsigned co = r / KHW;
  out[o] = w[(co * CIN + ci) * KHW + tap];
}

// tconv: one copy per parity class, flip baked in, rows padded to calloc:
// out[((cls*CALLOC + co)*4 + t)*CIN + ci] = co<cout ? w[(co*CIN+ci)*16+wtap] : 0
template <int CIN>
__global__ __launch_bounds__(256) void pack_tconv_w(
    const float* __restrict__ w, float* __restrict__ out, unsigned calloc_,
    unsigned cout, unsigned total) {
  const unsigned o = blockIdx.x * 256u + threadIdx.x;
  if (o >= total) return;
  const unsigned ci = o & (CIN - 1);
  unsigned r = o / CIN;
  const unsigned t = r & 3u; r >>= 2;
  const unsigned co = r % calloc_;
  const unsigned cls = r / calloc_;
  const unsigned py = cls >> 1, px = cls & 1u;
  const unsigned kyi = t >> 1, kxi = t & 1u;
  const unsigned wtap = (3u - py - 2u * kyi) * 4u + (3u - px - 2u * kxi);
  out[o] = (co < cout) ? w[(co * CIN + ci) * 16u + wtap] : 0.f;
}

// ---------------------------------------------------------------------------
// Forward convolution as implicit GEMM.  M = 32*HOUT*WOUT, N = COUT,
// K = CIN*KH*KW.  Wave: 16(M) x 16*NT(N); 8 waves/block.
// KC>0: TDM stages 16*NT x KC weight tiles into LDS (double-buffered,
//       row stride KC+2 via TDM padding -> conflict-free ds_load_b64).
// KC=0: direct global B loads (layer 1: NCHW input, K=48).
// ---------------------------------------------------------------------------
template <int CIN, int KH, int KW, int STRIDE, int PAD, int HIN, int WIN,
          int HOUT, int WOUT, int COUT, bool RELU, bool IN_NCHW, int NT, int KC>
__global__ __launch_bounds__(256) void conv_wmma(
    const float* __restrict__ in, const float* __restrict__ w,
    const float* __restrict__ bias, float* __restrict__ out) {
  constexpr int K = CIN * KH * KW;
  static_assert(KC == 0 || (K % KC) == 0, "KC must divide K");
  const unsigned lane = threadIdx.x & 31u;
  const unsigned wv = threadIdx.x >> 5;
  const unsigned half = lane >> 4;
  const unsigned l16 = lane & 15u;

  const unsigned m0 = (blockIdx.x * 8u + wv) * 16u;
  const unsigned nb = blockIdx.y * (16u * NT);

  const unsigned mA = m0 + l16;
  const unsigned ox = mA % WOUT;
  const unsigned oy = (mA / WOUT) % HOUT;
  const unsigned ni = mA / (WOUT * HOUT);
  const int iy0 = (int)oy * STRIDE - PAD;
  const int ix0 = (int)ox * STRIDE - PAD;

  // im2col A-fragment gather for global k index kk (lane supplies kk, kk+1)
  auto loadA = [&](unsigned kk) -> v2f {
    v2f a = {};
    if constexpr (IN_NCHW) {
#pragma unroll
      for (int u = 0; u < 2; ++u) {
        const unsigned k = kk + (unsigned)u;
        const unsigned ci = k / (KH * KW), tap = k % (KH * KW);
        const int iy = iy0 + (int)(tap / KW), ix = ix0 + (int)(tap % KW);
        if (iy >= 0 && iy < HIN && ix >= 0 && ix < WIN)
          a[u] = in[((ni * CIN + ci) * HIN + (unsigned)iy) * WIN + (unsigned)ix];
      }
    } else {
      const unsigned tap = kk / CIN, ci = kk % CIN;  // CIN power of two
      const int iy = iy0 + (int)(tap / KW), ix = ix0 + (int)(tap % KW);
      if (iy >= 0 && iy < HIN && ix >= 0 && ix < WIN)
        a = *(const v2f*)&in[((ni * HIN + (unsigned)iy) * WIN + (unsigned)ix) * CIN + ci];
    }
    return a;
  };

  v8f acc[NT];
#pragma unroll
  for (int t = 0; t < NT; ++t) acc[t] = (v8f){};

  if constexpr (KC > 0) {
    constexpr int NROW = 16 * NT;
    constexpr int RS = KC + 2;                    // padded LDS row stride
    constexpr unsigned PADI = __builtin_ctz(KC) - 1;  // interval code: KC dwords
    constexpr int NCH = K / KC;
    __shared__ float wl[2][NROW * RS];

    if (wv == 0) {  // wave-uniform: EXEC stays full for WMMA elsewhere
      tdm_load_2d(lds_off(&wl[0][0]), w + (size_t)nb * K, KC, NROW, K, PADI, 1, true);
      tdm_wait0();
    }
    __syncthreads();
    for (int ch = 0; ch < NCH; ++ch) {
      const unsigned buf = (unsigned)ch & 1u;
      if (wv == 0 && ch + 1 < NCH)
        tdm_load_2d(lds_off(&wl[buf ^ 1u][0]),
                    w + (size_t)nb * K + (size_t)(ch + 1) * KC, KC, NROW, K,
                    PADI, 1, true);
#pragma unroll 4
      for (int kc = 0; kc < KC; kc += 4) {
        const unsigned kk = (unsigned)(ch * KC + kc) + half * 2u;
        const v2f a = loadA(kk);
        const unsigned kl = (unsigned)kc + half * 2u;
#pragma unroll
        for (int t = 0; t < NT; ++t) {
          const v2f b = *(const v2f*)&wl[buf][((unsigned)t * 16u + l16) * RS + kl];
          acc[t] = wmma4(a, b, acc[t]);
        }
      }
      if (wv == 0 && ch + 1 < NCH) tdm_wait0();
      __syncthreads();
    }
  } else {
#pragma unroll 4
    for (int k0 = 0; k0 < K; k0 += 4) {
      const unsigned kk = (unsigned)k0 + half * 2u;
      const v2f a = loadA(kk);
#pragma unroll
      for (int t = 0; t < NT; ++t) {
        const unsigned co = nb + (unsigned)t * 16u + l16;
        const v2f b = *(const v2f*)&w[co * (unsigned)K + kk];
        acc[t] = wmma4(a, b, acc[t]);
      }
    }
  }

  // ---- epilogue: bias (+ReLU), NHWC store ----
#pragma unroll
  for (int t = 0; t < NT; ++t) {
#pragma unroll
    for (int r = 0; r < 8; ++r) {
      const unsigned m = m0 + (unsigned)r + half * 8u;
      const unsigned co = nb + (unsigned)t * 16u + l16;
      const unsigned oxx = m % WOUT;
      const unsigned oyy = (m / WOUT) % HOUT;
      const unsigned nii = m / (WOUT * HOUT);
      float v = acc[t][r] + bias[co];
      if (RELU) v = fmaxf(v, 0.f);
      out[((nii * HOUT + oyy) * WOUT + oxx) * COUT + co] = v;
    }
  }
}

// ---------------------------------------------------------------------------
// Transposed conv (k=4,s=2,p=1) via parity classes (blockIdx.z = py*2+px):
//   oy=2*ty+py, ox=2*tx+px; taps ky in {py,py+2}, kx in {px,px+2};
//   input coords (ty+py-1+kyi, tx+px-1+kxi); K=4*CIN, k = t*CIN + ci.
// Weights pre-packed per class (flip baked in, CALLOC rows incl. zero pad);
// TDM-staged through LDS like conv_wmma.
// ---------------------------------------------------------------------------
template <int CIN, int COUT, int CALLOC, int HIN, int WIN, int HT, int WT,
          bool RELU, bool OUT_NCHW, int NT, int KC>
__global__ __launch_bounds__(256) void tconv_wmma(
    const float* __restrict__ in, const float* __restrict__ wpk,
    const float* __restrict__ bias, float* __restrict__ out) {
  constexpr int K = 4 * CIN;
  static_assert((K % KC) == 0, "KC must divide K");
  constexpr int NROW = 16 * NT;
  constexpr int RS = KC + 2;
  constexpr unsigned PADI = __builtin_ctz(KC) - 1;
  constexpr int NCH = K / KC;
  __shared__ float wl[2][NROW * RS];

  const unsigned lane = threadIdx.x & 31u;
  const unsigned wv = threadIdx.x >> 5;
  const unsigned half = lane >> 4;
  const unsigned l16 = lane & 15u;
  const unsigned cls = blockIdx.z;
  const unsigned py = cls >> 1, px = cls & 1u;
  const float* __restrict__ w = wpk + (size_t)cls * CALLOC * K;

  const unsigned m0 = (blockIdx.x * 8u + wv) * 16u;
  const unsigned nb = blockIdx.y * (16u * NT);

  const unsigned mA = m0 + l16;
  const unsigned tx = mA % WT;
  const unsigned ty = (mA / WT) % HT;
  const unsigned ni = mA / (WT * HT);
  const int iyb = (int)(ty + py) - 1, ixb = (int)(tx + px) - 1;

  v8f acc[NT];
#pragma unroll
  for (int t = 0; t < NT; ++t) acc[t] = (v8f){};

  if (wv == 0) {
    tdm_load_2d(lds_off(&wl[0][0]), w + (size_t)nb * K, KC, NROW, K, PADI, 1, true);
    tdm_wait0();
  }
  __syncthreads();
  for (int ch = 0; ch < NCH; ++ch) {
    const unsigned buf = (unsigned)ch & 1u;
    if (wv == 0 && ch + 1 < NCH)
      tdm_load_2d(lds_off(&wl[buf ^ 1u][0]),
                  w + (size_t)nb * K + (size_t)(ch + 1) * KC, KC, NROW, K,
                  PADI, 1, true);
#pragma unroll 4
    for (int kc = 0; kc < KC; kc += 4) {
      const unsigned kk = (unsigned)(ch * KC + kc) + half * 2u;
      const unsigned t4 = kk / CIN, ci = kk % CIN;
      const int iy = iyb + (int)(t4 >> 1), ix = ixb + (int)(t4 & 1u);
      v2f a = {};
      if (iy >= 0 && iy < HIN && ix >= 0 && ix < WIN)
        a = *(const v2f*)&in[((ni * HIN + (unsigned)iy) * WIN + (unsigned)ix) * CIN + ci];
      const unsigned kl = (unsigned)kc + half * 2u;
#pragma unroll
      for (int t = 0; t < NT; ++t) {
        const v2f b = *(const v2f*)&wl[buf][((unsigned)t * 16u + l16) * RS + kl];
        acc[t] = wmma4(a, b, acc[t]);
      }
    }
    if (wv == 0 && ch + 1 < NCH) tdm_wait0();
    __syncthreads();
  }

#pragma unroll
  for (int t = 0; t < NT; ++t) {
#pragma unroll
    for (int r = 0; r < 8; ++r) {
      const unsigned m = m0 + (unsigned)r + half * 8u;
      const unsigned co = nb + (unsigned)t * 16u + l16;
      const unsigned txx = m % WT;
      const unsigned tyy = (m / WT) % HT;
      const unsigned nii = m / (WT * HT);
      const unsigned oy = 2u * tyy + py, ox = 2u * txx + px;
      if (co < COUT) {
        float v = acc[t][r] + bias[co];
        if (RELU) v = fmaxf(v, 0.f);
        if (OUT_NCHW)
          out[((nii * COUT + co) * (2u * HT) + oy) * (2u * WT) + ox] = v;
        else
          out[((nii * (2u * HT) + oy) * (2u * WT) + ox) * COUT + co] = v;
      }
    }
  }
}

// ---------------------------------------------------------------------------
// Vector quantization.  z = [131072, 64] (NHWC).  Codebook chunks (128x64,
// 32KB) staged into LDS by TDM.  d = |z|^2+|c|^2-2 z.c; argmin; loss uses
// exact |z-c*|^2.  Deterministic fixed-order reductions.
// vq_loss = 1.25 * sum|z-zq|^2 / numel.
// ---------------------------------------------------------------------------
__global__ __launch_bounds__(256) void vq_kernel(
    const float* __restrict__ z, const float* __restrict__ cb,
    float* __restrict__ zq, float* __restrict__ idx_out,
    float* __restrict__ partial) {
  __shared__ float scb[128 * 64];
  __shared__ float scc[128];
  __shared__ float sred[256];
  const unsigned tid = threadIdx.x;
  const unsigned row = blockIdx.x * 256u + tid;

  float zr[64];
#pragma unroll
  for (int j = 0; j < 64; ++j) zr[j] = z[row * 64u + j];
  float zz = 0.f;
#pragma unroll
  for (int j = 0; j < 64; ++j) zz += zr[j] * zr[j];

  float bestd = 3.4e38f;
  unsigned best = 0;
  for (unsigned c0 = 0; c0 < 512; c0 += 128) {
    __syncthreads();  // previous chunk's readers done
    if (tid < 32) {   // wave 0 drives the DMA
      tdm_load_2d(lds_off(scb), cb + c0 * 64u, 64u, 128u, 64u, 0u, 0u, false);
      tdm_wait0();
    }
    __syncthreads();
    if (tid < 128) {
      float s = 0.f;
#pragma unroll
      for (int j = 0; j < 64; ++j) s += scb[tid * 64u + j] * scb[tid * 64u + j];
      scc[tid] = s;
    }
    __syncthreads();
    for (unsigned c = 0; c < 128; ++c) {
      float dot = 0.f;
#pragma unroll
      for (int j = 0; j < 64; ++j) dot += zr[j] * scb[c * 64u + j];
      const float d = zz + scc[c] - 2.f * dot;
      if (d < bestd) { bestd = d; best = c0 + c; }
    }
  }

  const float* crow = &cb[best * 64u];
  float err = 0.f;
#pragma unroll
  for (int j = 0; j < 64; ++j) {
    const float cv = crow[j];
    zq[row * 64u + j] = cv;
    const float dd = zr[j] - cv;
    err += dd * dd;
  }
  idx_out[row] = (float)best;

  sred[tid] = err;
  __syncthreads();
  for (unsigned s = 128; s > 0; s >>= 1) {
    if (tid < s) sred[tid] += sred[tid + s];
    __syncthreads();
  }
  if (tid == 0) partial[blockIdx.x] = sred[0];
}

__global__ __launch_bounds__(512) void vq_finalize(
    const float* __restrict__ partial, float* __restrict__ loss_out,
    float scale) {
  __shared__ float s[512];
  const unsigned tid = threadIdx.x;
  s[tid] = partial[tid];  // exactly 512 partials
  __syncthreads();
  for (unsigned st = 256; st > 0; st >>= 1) {
    if (tid < st) s[tid] += s[tid + st];
    __syncthreads();
  }
  if (tid == 0) *loss_out = s[0] * scale;
}

// ---------------------------------------------------------------------------
// Host-side orchestration.
// ---------------------------------------------------------------------------
extern "C" void kernel_launch(void* const* d_in, const int* in_sizes, int n_in,
                              void* d_out, int out_size, void* d_ws,
                              size_t ws_size, hipStream_t stream) {
  const float* x   = (const float*)d_in[0];   // [32,3,256,256] NCHW
  const float* ew1 = (const float*)d_in[1];   // [512,3,4,4]
  const float* eb1 = (const float*)d_in[2];
  const float* ew2 = (const float*)d_in[3];   // [512,512,4,4]
  const float* eb2 = (const float*)d_in[4];
  const float* ew3 = (const float*)d_in[5];   // [64,512,3,3]
  const float* eb3 = (const float*)d_in[6];
  const float* cbk = (const float*)d_in[7];   // [512,64]
  const float* dw1 = (const float*)d_in[8];   // [512,64,3,3]
  const float* db1 = (const float*)d_in[9];
  const float* tw1 = (const float*)d_in[10];  // [512,512,4,4]
  const float* tb1 = (const float*)d_in[11];
  const float* tw2 = (const float*)d_in[12];  // [3,512,4,4]
  const float* tb2 = (const float*)d_in[13];

  // Workspace layout (floats).  h1/g2 and h2/g1 overlap (disjoint lifetimes).
  float* ws = (float*)d_ws;
  float* h1 = ws;                         // [32,128,128,512] NHWC (1 GB)
  float* h2 = h1 + 268435456LL;           // [32, 64, 64,512] NHWC (256 MB)
  float* z  = h2 + 67108864LL;            // [32, 64, 64, 64] == [131072,64]
  float* zq = z + 8388608LL;
  float* part = zq + 8388608LL;           // 512 partial loss sums
  float* pk_ew2 = part + 512;             // 512*8192
  float* pk_ew3 = pk_ew2 + 4194304LL;     // 64*4608
  float* pk_dw1 = pk_ew3 + 294912LL;      // 512*576
  float* pk_tw1 = pk_dw1 + 294912LL;      // 4*512*2048
  float* pk_tw2 = pk_tw1 + 4194304LL;     // 4*64*2048 (rows zero-padded 3->64)
  float* g1 = h2;                         // reuse (h2 dead after conv3)
  float* g2 = h1;                         // reuse (h1 dead after conv2)

  // d_out: x_recon [32,3,256,256] ++ vq_loss [1] ++ indices [131072]
  float* xrec = (float*)d_out;
  float* loss = xrec + 6291456LL;
  float* idxf = loss + 1;

  const dim3 blk(256);

  // ---- weight repacks (tiny, one pass each) ----
  pack_conv_w<512, 16><<<dim3(4194304u / 256), blk, 0, stream>>>(ew2, pk_ew2, 4194304u);
  pack_conv_w<512, 9><<<dim3(1152), blk, 0, stream>>>(ew3, pk_ew3, 294912u);
  pack_conv_w<64, 9><<<dim3(1152), blk, 0, stream>>>(dw1, pk_dw1, 294912u);
  pack_tconv_w<512><<<dim3(4194304u / 256), blk, 0, stream>>>(tw1, pk_tw1, 512u, 512u, 4194304u);
  pack_tconv_w<512><<<dim3(524288u / 256), blk, 0, stream>>>(tw2, pk_tw2, 64u, 3u, 524288u);

  // ---- encoder ----
  // conv1: 3->512 k4 s2 p1 + ReLU.  M=524288 -> x=4096; direct B loads (K=48)
  conv_wmma<3, 4, 4, 2, 1, 256, 256, 128, 128, 512, true, true, 4, 0>
      <<<dim3(4096, 8, 1), blk, 0, stream>>>(x, ew1, eb1, h1);
  // conv2: 512->512 k4 s2 p1 + ReLU.  K=8192, TDM-staged KC=64
  conv_wmma<512, 4, 4, 2, 1, 128, 128, 64, 64, 512, true, false, 4, 64>
      <<<dim3(1024, 8, 1), blk, 0, stream>>>(h1, pk_ew2, eb2, h2);
  // conv3: 512->64 k3 s1 p1 -> z.  K=4608
  conv_wmma<512, 3, 3, 1, 1, 64, 64, 64, 64, 64, false, false, 4, 64>
      <<<dim3(1024, 1, 1), blk, 0, stream>>>(h2, pk_ew3, eb3, z);

  // ---- vector quantization + loss ----
  vq_kernel<<<dim3(512, 1, 1), blk, 0, stream>>>(z, cbk, zq, idxf, part);
  vq_finalize<<<dim3(1, 1, 1), dim3(512), 0, stream>>>(
      part, loss, 1.25f / (131072.f * 64.f));

  // ---- decoder (forward value of straight-through z_q_st == z_q) ----
  conv_wmma<64, 3, 3, 1, 1, 64, 64, 64, 64, 512, true, false, 4, 64>
      <<<dim3(1024, 8, 1), blk, 0, stream>>>(zq, pk_dw1, db1, g1);
  tconv_wmma<512, 512, 512, 64, 64, 64, 64, true, false, 4, 64>
      <<<dim3(1024, 8, 4), blk, 0, stream>>>(g1, pk_tw1, tb1, g2);
  tconv_wmma<512, 3, 64, 128, 128, 128, 128, false, true, 1, 64>
      <<<dim3(4096, 1, 4), blk, 0, stream>>>(g2, pk_tw2, tb2, xrec);
}